// TextProcessor_BiLSTM_5703716569306
// MI455X (gfx1250) — compile-verified
//
#include <hip/hip_runtime.h>
#include <math.h>

// Problem constants (from reference)
#define DD   200
#define DP   224            // D padded to 7*32 for uniform K-chunking
#define HH   1024
#define BB   64
#define QQ   8
#define TT   64
#define NN   (BB*QQ)        // 512 sequences
#define G3   (3*HH)         // 3072

#define KC        32        // K-chunk staged per pipeline stage
#define NCH_REC   (HH/KC)   // 32 recurrent chunks
#define NCH_IN    (DP/KC)   // 7 input-projection chunks
#define NCHUNK    (NCH_REC + NCH_IN)   // 39
#define ASTRIDE   36        // padded LDS row stride for A (16B aligned, bank-spread)

typedef float v2f __attribute__((ext_vector_type(2)));
typedef float v8f __attribute__((ext_vector_type(8)));

__device__ __forceinline__ float sigmoid_(float x) {
    return 1.0f / (1.0f + __expf(-x));
}

// ---------------------------------------------------------------------------
// Kernel 1: embedding gather (padded to DP) + Masking(0.0) byte per token
// ---------------------------------------------------------------------------
__global__ void prep_kernel(const int* __restrict__ x,
                            const float* __restrict__ table,
                            float* __restrict__ embp,
                            unsigned char* __restrict__ mask) {
    __shared__ int s_nz;
    const int token = blockIdx.x;          // n*TT + t
    const int d = threadIdx.x;
    if (d == 0) s_nz = 0;
    __syncthreads();
    const int idx = x[token];
    if (d < DP) {
        float v = (d < DD) ? table[(size_t)idx * DD + d] : 0.0f;
        embp[(size_t)token * DP + d] = v;
        if (v != 0.0f) atomicOr(&s_nz, 1);
    }
    __syncthreads();
    if (d == 0) mask[token] = (unsigned char)(s_nz != 0);
}

// ---------------------------------------------------------------------------
// Kernel 2: copy input kernel [200x3072] into zero-padded [224x3072]
// ---------------------------------------------------------------------------
__global__ void padw_kernel(const float* __restrict__ src, float* __restrict__ dst) {
    int i = blockIdx.x * blockDim.x + threadIdx.x;
    if (i < DP * G3) {
        int row = i / G3;
        dst[i] = (row < DD) ? src[i] : 0.0f;
    }
}

// ---------------------------------------------------------------------------
// Kernel 3: zero initial hidden state (ping buffers)
// ---------------------------------------------------------------------------
__global__ void zero_kernel(float* __restrict__ a, float* __restrict__ b, int n) {
    int i = blockIdx.x * blockDim.x + threadIdx.x;
    if (i < n) { a[i] = 0.0f; b[i] = 0.0f; }
}

// ---------------------------------------------------------------------------
// Kernel 4: one GRU time step, fwd+bwd (blockIdx.z), 64(M)x64(N) block tile.
// Double-buffered async staging (global_load_async_to_lds) -> ds_load_b64
// fragments -> v_wmma_f32_16x16x4_f32. Two phase loops (recurrent K=1024,
// input-projection K=224) so each loop has a fixed accumulator set.
// ---------------------------------------------------------------------------
__global__ __launch_bounds__(512)
void gru_step_kernel(const float* __restrict__ embp,
                     const unsigned char* __restrict__ mask,
                     const float* __restrict__ hf_in, float* __restrict__ hf_out,
                     const float* __restrict__ hb_in, float* __restrict__ hb_out,
                     const float* __restrict__ wx_f, const float* __restrict__ rk_f,
                     const float* __restrict__ b_f,
                     const float* __restrict__ wx_b, const float* __restrict__ rk_b,
                     const float* __restrict__ b_b,
                     float* __restrict__ out, int s)
{
    // A tiles: 64 rows x KC, padded row stride (2 buffers)
    __shared__ __align__(16) float Asm[2][64 * ASTRIDE];
    // B tiles: 3 gates x KC x 64 cols in fragment order:
    //   off = gate*2048 + (k>>2)*256 + col*4 + ((k>>1)&1)*2 + (k&1)   (floats)
    __shared__ __align__(16) float Bsm[2][3 * KC * 64];

    const int dir = blockIdx.z;
    const int p = dir ? (TT - 1 - s) : s;
    const float* __restrict__ hin  = dir ? hb_in  : hf_in;
    float*       __restrict__ hout = dir ? hb_out : hf_out;
    const float* __restrict__ Wx   = dir ? wx_b : wx_f;    // [224 x 3072] padded
    const float* __restrict__ Wr   = dir ? rk_b : rk_f;    // [1024 x 3072]
    const float* __restrict__ bias = dir ? b_b  : b_f;     // [2 x 3072]

    const int tid    = threadIdx.x;
    const int wave   = tid >> 5;
    const int lane   = tid & 31;
    const int laneN  = lane & 15;
    const int laneHi = lane >> 4;

    const int colBlock = blockIdx.x * 64;                 // hidden-unit tile base
    const int rowBlock = blockIdx.y * 64;                 // sequence tile base
    const int colLocal = (wave & 3) * 16 + laneN;         // 0..63
    const int rowWave  = (wave >> 2) * 16;                // 0,16,32,48

    // --- per-thread staging maps (B: 12 b32 asyncs; A: 1 b128 async) ---
    int goffB[12];       // element offset inside (kBase,colBlock) weight tile
    int loffB[12];       // byte offset inside one B LDS buffer
    #pragma unroll
    for (int j = 0; j < 12; ++j) {
        const int li = j * 512 + tid;                     // 0..6143
        const int c  = li & 63;
        const int kr = (li >> 6) & 31;
        const int g  = li >> 11;                          // 0..2
        goffB[j] = kr * G3 + g * HH + c;
        loffB[j] = 4 * (g * 2048 + (kr >> 2) * 256 + c * 4 + ((kr >> 1) & 1) * 2 + (kr & 1));
    }
    const int rowS = tid >> 3;                            // 0..63  (A staging row)
    const int segS = tid & 7;                             // 0..7   (16B segment)

    const unsigned ldsB[2] = { (unsigned)(size_t)&Bsm[0][0], (unsigned)(size_t)&Bsm[1][0] };
    const unsigned ldsA[2] = { (unsigned)(size_t)&Asm[0][0], (unsigned)(size_t)&Asm[1][0] };
    const unsigned loffA = (unsigned)(rowS * ASTRIDE * 4 + segS * 16);

    // Issue one pipeline stage: chunk index c into LDS buffer buf.
    auto stage = [&](int c, int buf) {
        const bool rec = (c < NCH_REC);
        const int kB = rec ? c * KC : (c - NCH_REC) * KC;
        const float* Wsrc = rec ? Wr : Wx;
        const unsigned long long bbase =
            (unsigned long long)(size_t)Wsrc + (unsigned long long)(kB * G3 + colBlock) * 4ull;
        const unsigned lB = ldsB[buf];
        #pragma unroll
        for (int j = 0; j < 12; ++j) {
            unsigned long long ga = bbase + (unsigned long long)((unsigned)goffB[j] * 4u);
            unsigned la = lB + (unsigned)loffB[j];
            asm volatile("global_load_async_to_lds_b32 %0, %1, off"
                         :: "v"(la), "v"(ga) : "memory");
        }
        unsigned long long ab;
        if (rec) {
            ab = (unsigned long long)(size_t)hin
               + ((unsigned long long)(rowBlock + rowS) * HH + kB) * 4ull + segS * 16ull;
        } else {
            ab = (unsigned long long)(size_t)embp
               + (((unsigned long long)(rowBlock + rowS) * TT + p) * DP + kB) * 4ull + segS * 16ull;
        }
        unsigned laA = ldsA[buf] + loffA;
        asm volatile("global_load_async_to_lds_b128 %0, %1, off"
                     :: "v"(laA), "v"(ab) : "memory");
    };

    v8f accZ  = {0.f,0.f,0.f,0.f,0.f,0.f,0.f,0.f};
    v8f accR  = accZ;
    v8f accXH = accZ;
    v8f accRH = accZ;

    const int aOff = (rowWave + laneN) * ASTRIDE + 2 * laneHi;   // floats
    const int bOff = colLocal * 4 + laneHi * 2;                  // floats

    stage(0, 0);

    // ---- phase 1: recurrent GEMMs (chunks 0..31), third acc = accRH ----
    for (int c = 0; c < NCH_REC; ++c) {
        const int buf = c & 1;
        stage(c + 1, buf ^ 1);                               // c+1 <= 32 always exists
        asm volatile("s_wait_asynccnt 0xd" ::: "memory");    // newest 13 still in flight
        __syncthreads();
        const float* As = &Asm[buf][0];
        const float* Bs = &Bsm[buf][0];
        #pragma unroll
        for (int kk = 0; kk < KC; kk += 4) {
            v2f a  = *(const v2f*)(As + aOff + kk);
            v2f bz = *(const v2f*)(Bs + 0 * 2048 + kk * 64 + bOff);
            v2f br = *(const v2f*)(Bs + 1 * 2048 + kk * 64 + bOff);
            v2f bh = *(const v2f*)(Bs + 2 * 2048 + kk * 64 + bOff);
            accZ  = __builtin_amdgcn_wmma_f32_16x16x4_f32(false, a, false, bz, (short)0, accZ,  false, false);
            accR  = __builtin_amdgcn_wmma_f32_16x16x4_f32(false, a, false, br, (short)0, accR,  false, false);
            accRH = __builtin_amdgcn_wmma_f32_16x16x4_f32(false, a, false, bh, (short)0, accRH, false, false);
        }
        __syncthreads();   // all waves done reading before buffer is re-staged
    }

    // ---- phase 2: input-projection GEMMs (chunks 32..38), third acc = accXH ----
    for (int c = NCH_REC; c < NCHUNK; ++c) {
        const int buf = c & 1;
        if (c + 1 < NCHUNK) {
            stage(c + 1, buf ^ 1);
            asm volatile("s_wait_asynccnt 0xd" ::: "memory");
        } else {
            asm volatile("s_wait_asynccnt 0x0" ::: "memory");
        }
        __syncthreads();
        const float* As = &Asm[buf][0];
        const float* Bs = &Bsm[buf][0];
        #pragma unroll
        for (int kk = 0; kk < KC; kk += 4) {
            v2f a  = *(const v2f*)(As + aOff + kk);
            v2f bz = *(const v2f*)(Bs + 0 * 2048 + kk * 64 + bOff);
            v2f br = *(const v2f*)(Bs + 1 * 2048 + kk * 64 + bOff);
            v2f bh = *(const v2f*)(Bs + 2 * 2048 + kk * 64 + bOff);
            accZ  = __builtin_amdgcn_wmma_f32_16x16x4_f32(false, a, false, bz, (short)0, accZ,  false, false);
            accR  = __builtin_amdgcn_wmma_f32_16x16x4_f32(false, a, false, br, (short)0, accR,  false, false);
            accXH = __builtin_amdgcn_wmma_f32_16x16x4_f32(false, a, false, bh, (short)0, accXH, false, false);
        }
        __syncthreads();
    }

    // ---- epilogue: gates, mask carry, state + output writes ----
    const int col = colBlock + colLocal;
    const float bz_s = bias[col]           + bias[G3 + col];
    const float br_s = bias[HH + col]      + bias[G3 + HH + col];
    const float bxh  = bias[2 * HH + col];
    const float brh  = bias[G3 + 2 * HH + col];
    const int mrow0 = rowBlock + rowWave + laneHi * 8;

    #pragma unroll
    for (int r = 0; r < 8; ++r) {
        const int n = mrow0 + r;
        const float hprev = hin[(size_t)n * HH + col];
        const float z  = sigmoid_(accZ[r] + bz_s);
        const float rg = sigmoid_(accR[r] + br_s);
        const float hh = tanhf(accXH[r] + bxh + rg * (accRH[r] + brh));
        const float hn = z * hprev + (1.0f - z) * hh;
        const float ho = mask[n * TT + p] ? hn : hprev;
        hout[(size_t)n * HH + col] = ho;
        const size_t oi = ((size_t)n * TT + p) * HH + col;
        const float val = 0.5f * ho;
        // First-arriving direction (launch s<32) stores; the other accumulates.
        if (s < TT / 2) out[oi] = val;
        else            out[oi] += val;
    }
}

// ---------------------------------------------------------------------------
// Host launcher
// ---------------------------------------------------------------------------
extern "C" void kernel_launch(void* const* d_in, const int* in_sizes, int n_in,
                              void* d_out, int out_size, void* d_ws, size_t ws_size,
                              hipStream_t stream)
{
    const int*   x     = (const int*)  d_in[0];
    const float* table = (const float*)d_in[1];
    const float* k_f   = (const float*)d_in[2];
    const float* rk_f  = (const float*)d_in[3];
    const float* b_f   = (const float*)d_in[4];
    const float* k_b   = (const float*)d_in[5];
    const float* rk_b  = (const float*)d_in[6];
    const float* b_b   = (const float*)d_in[7];
    float* out = (float*)d_out;

    // Workspace carving (~43 MB)
    char* w = (char*)d_ws;
    size_t off = 0;
    float* embp  = (float*)(w + off); off += (size_t)NN * TT * DP * sizeof(float);
    float* wxp_f = (float*)(w + off); off += (size_t)DP * G3 * sizeof(float);
    float* wxp_b = (float*)(w + off); off += (size_t)DP * G3 * sizeof(float);
    float* hf0   = (float*)(w + off); off += (size_t)NN * HH * sizeof(float);
    float* hf1   = (float*)(w + off); off += (size_t)NN * HH * sizeof(float);
    float* hb0   = (float*)(w + off); off += (size_t)NN * HH * sizeof(float);
    float* hb1   = (float*)(w + off); off += (size_t)NN * HH * sizeof(float);
    unsigned char* mask = (unsigned char*)(w + off); off += (size_t)NN * TT;

    // 1) gather embeddings (padded) + mask
    prep_kernel<<<NN * TT, 256, 0, stream>>>(x, table, embp, mask);

    // 2) zero-padded input kernels
    {
        const int n = DP * G3;
        padw_kernel<<<(n + 255) / 256, 256, 0, stream>>>(k_f, wxp_f);
        padw_kernel<<<(n + 255) / 256, 256, 0, stream>>>(k_b, wxp_b);
    }

    // 3) zero initial hidden states
    {
        const int n = NN * HH;
        zero_kernel<<<(n + 255) / 256, 256, 0, stream>>>(hf0, hb0, n);
    }

    // 4) 64 sequential GRU steps, fwd+bwd fused, ping-pong h state
    float* hfr = hf0; float* hfw = hf1;
    float* hbr = hb0; float* hbw = hb1;
    for (int s = 0; s < TT; ++s) {
        gru_step_kernel<<<dim3(HH / 64, NN / 64, 2), 512, 0, stream>>>(
            embp, mask, hfr, hfw, hbr, hbw,
            wxp_f, rk_f, b_f, wxp_b, rk_b, b_b, out, s);
        float* t;
        t = hfr; hfr = hfw; hfw = t;
        t = hbr; hbr = hbw; hbw = t;
    }
}